// LoRAElementLinear_17600775979361
// MI455X (gfx1250) — compile-verified
//
#include <hip/hip_runtime.h>
#include <hip/hip_bf16.h>

// Problem constants (match reference)
#define B_NODES 8192
#define Z_DIM   10
#define IN_D    512
#define OUT_D   512
#define R_DIM   8
#define M_DIM   3
#define SCALING 1.0f                    // LORA_ALPHA / R = 8/8
#define ALPHA_S 0.04419417382415922f    // 1/sqrt(512)

// GEMM tiling
#define NODES_PER_BLK 16
#define N_TILE  (NODES_PER_BLK * M_DIM)   // 48 columns
#define M_TILE  128                        // 8 waves * 16 rows
#define K_CHUNK 64
#define BS_STRIDE 66                       // padded LDS stride (even -> 8B aligned b64)
#define MAX_DESC 544                       // >= ceil(8192/16) + Z

typedef float v2f __attribute__((ext_vector_type(2)));
typedef float v8f __attribute__((ext_vector_type(8)));

// ---------------- workspace layout (in d_ws) ----------------
// [0]                      Wm      : Z*OUT*IN floats (merged weights, 10 MB)
// then (ints):             zidx    : B
//                          sorted  : B
//                          ctl     : 64 ints (counts[16], offs[16], cursors[16], ndesc, pad)
//                          descs   : 3*MAX_DESC ints (z, start, nnodes)
#define WM_ELEMS ((size_t)Z_DIM * OUT_D * IN_D)

__global__ void k_zero_ctl(int* ctl) {
    ctl[threadIdx.x] = 0;   // 64 threads zero counts/offs/cursors/ndesc
}

__global__ void k_classify(const float* __restrict__ na, int* __restrict__ zidx,
                           int* __restrict__ counts) {
    int b = blockIdx.x * 256 + threadIdx.x;
    if (b >= B_NODES) return;
    int z = 0;
    #pragma unroll
    for (int j = 0; j < Z_DIM; ++j)
        if (na[b * Z_DIM + j] > 0.5f) z = j;
    zidx[b] = z;
    atomicAdd(&counts[z], 1);
}

// single-thread scan + descriptor build (Z=10, trivial)
__global__ void k_scan_desc(int* __restrict__ ctl, int* __restrict__ descs) {
    if (threadIdx.x != 0 || blockIdx.x != 0) return;
    int* counts  = ctl;
    int* offs    = ctl + 16;
    int* ndesc   = ctl + 48;
    int running = 0;
    for (int z = 0; z < Z_DIM; ++z) { offs[z] = running; running += counts[z]; }
    int nd = 0;
    for (int z = 0; z < Z_DIM; ++z) {
        int cnt = counts[z];
        for (int j = 0; j < cnt; j += NODES_PER_BLK) {
            int rem = cnt - j;
            descs[3 * nd + 0] = z;
            descs[3 * nd + 1] = offs[z] + j;
            descs[3 * nd + 2] = rem < NODES_PER_BLK ? rem : NODES_PER_BLK;
            ++nd;
        }
    }
    *ndesc = nd;
}

__global__ void k_scatter(const int* __restrict__ zidx, int* __restrict__ ctl,
                          int* __restrict__ sorted) {
    int b = blockIdx.x * 256 + threadIdx.x;
    if (b >= B_NODES) return;
    int z = zidx[b];
    int pos = ctl[16 + z] /*offs*/ + atomicAdd(&ctl[32 + z] /*cursor*/, 1);
    sorted[pos] = b;
}

// Wm[z,o,i] = (weights[z,o,i] + SCALING * sum_r lora_A[z,r,i]*lora_B[z,o,r]) * ALPHA
__global__ void k_merge(const float* __restrict__ w, const float* __restrict__ lA,
                        const float* __restrict__ lB, float* __restrict__ Wm) {
    int idx = blockIdx.x * 256 + threadIdx.x;          // exact grid, Z*OUT*IN
    int z   = idx / (OUT_D * IN_D);
    int rem = idx - z * (OUT_D * IN_D);
    int o   = rem / IN_D;
    int i   = rem - o * IN_D;
    float acc = 0.0f;
    #pragma unroll
    for (int r = 0; r < R_DIM; ++r)
        acc = fmaf(lA[(z * R_DIM + r) * IN_D + i], lB[(z * OUT_D + o) * R_DIM + r], acc);
    Wm[idx] = (w[idx] + SCALING * acc) * ALPHA_S;
}

// Main GEMM: per block, one node-group descriptor x one 128-row slice.
// out[node, C, d] = sum_c Wm[z, C, c] * t[node, c, d]
__global__ __launch_bounds__(256) void k_gemm(const float* __restrict__ t,
                                              const float* __restrict__ Wm,
                                              const int* __restrict__ sorted,
                                              const int* __restrict__ ctl,
                                              const int* __restrict__ descs,
                                              float* __restrict__ out) {
    __shared__ float Bs[N_TILE * BS_STRIDE];   // transposed: Bs[col][k], padded stride

    const int db = blockIdx.x;
    if (db >= ctl[48] /*ndesc*/) return;       // uniform branch

    const int z     = descs[3 * db + 0];
    const int start = descs[3 * db + 1];
    const int nnod  = descs[3 * db + 2];
    const int ncols = nnod * M_DIM;

    const int wave = threadIdx.x >> 5;
    const int lane = threadIdx.x & 31;
    const int lm   = lane & 15;
    const int kh   = lane >> 4;
    const int row0 = blockIdx.y * M_TILE + wave * 16;

    const float* Wz = Wm + (size_t)z * (OUT_D * IN_D);

    v8f acc[3];
    acc[0] = (v8f){}; acc[1] = (v8f){}; acc[2] = (v8f){};

    for (int kbase = 0; kbase < IN_D; kbase += K_CHUNK) {
        __syncthreads();
        // gather B tile: Bs[col][kk] = t[node(col), kbase+kk, col%3]  (zeros past ncols)
        for (int idx = threadIdx.x; idx < N_TILE * K_CHUNK; idx += 256) {
            int col = idx >> 6;         // / K_CHUNK
            int kk  = idx & (K_CHUNK - 1);
            float v = 0.0f;
            if (col < ncols) {
                int ni = col / 3;
                int d  = col - ni * 3;
                int node = sorted[start + ni];
                v = t[(size_t)node * (IN_D * M_DIM) + (size_t)(kbase + kk) * M_DIM + d];
            }
            Bs[col * BS_STRIDE + kk] = v;
        }
        __syncthreads();

        // A row pointer for this lane: row = row0+lm, cols kbase+2*kh + ks
        const float* wp = Wz + (size_t)(row0 + lm) * IN_D + kbase + 2 * kh;
        #pragma unroll
        for (int ks = 0; ks < K_CHUNK; ks += 4) {
            v2f a  = *(const v2f*)(wp + ks);
            v2f b0 = *(const v2f*)(&Bs[(0 * 16 + lm) * BS_STRIDE + ks + 2 * kh]);
            v2f b1 = *(const v2f*)(&Bs[(1 * 16 + lm) * BS_STRIDE + ks + 2 * kh]);
            v2f b2 = *(const v2f*)(&Bs[(2 * 16 + lm) * BS_STRIDE + ks + 2 * kh]);
            acc[0] = __builtin_amdgcn_wmma_f32_16x16x4_f32(false, a, false, b0,
                                                           (short)0, acc[0], false, false);
            acc[1] = __builtin_amdgcn_wmma_f32_16x16x4_f32(false, a, false, b1,
                                                           (short)0, acc[1], false, false);
            acc[2] = __builtin_amdgcn_wmma_f32_16x16x4_f32(false, a, false, b2,
                                                           (short)0, acc[2], false, false);
        }
    }

    // Writeback: C/D layout -> vgpr r, lanes0-15: M=r, lanes16-31: M=8+r, N=lane&15
    const int mbase = row0 + 8 * kh;
    #pragma unroll
    for (int nsub = 0; nsub < 3; ++nsub) {
        int col = nsub * 16 + lm;
        if (col < ncols) {
            int ni = col / 3;
            int d  = col - ni * 3;
            int node = sorted[start + ni];
            float* op = out + (size_t)node * (OUT_D * M_DIM) + d;
            #pragma unroll
            for (int r = 0; r < 8; ++r)
                op[(size_t)(mbase + r) * M_DIM] = acc[nsub][r];
        }
    }
}

extern "C" void kernel_launch(void* const* d_in, const int* in_sizes, int n_in,
                              void* d_out, int out_size, void* d_ws, size_t ws_size,
                              hipStream_t stream) {
    const float* t   = (const float*)d_in[0];   // [B, IN, M]
    const float* na  = (const float*)d_in[1];   // [B, Z]
    const float* w   = (const float*)d_in[2];   // [Z, OUT, IN]
    const float* lA  = (const float*)d_in[3];   // [Z, R, IN]
    const float* lB  = (const float*)d_in[4];   // [Z, OUT, R]
    float* out = (float*)d_out;                 // [B, OUT, M]

    float* Wm     = (float*)d_ws;
    int*   zidx   = (int*)(Wm + WM_ELEMS);
    int*   sorted = zidx + B_NODES;
    int*   ctl    = sorted + B_NODES;           // counts[16], offs[16], cursors[16], ndesc
    int*   descs  = ctl + 64;
    int*   counts = ctl;

    hipLaunchKernelGGL(k_zero_ctl, dim3(1), dim3(64), 0, stream, ctl);
    hipLaunchKernelGGL(k_classify, dim3(B_NODES / 256), dim3(256), 0, stream,
                       na, zidx, counts);
    hipLaunchKernelGGL(k_scan_desc, dim3(1), dim3(32), 0, stream, ctl, descs);
    hipLaunchKernelGGL(k_scatter, dim3(B_NODES / 256), dim3(256), 0, stream,
                       zidx, ctl, sorted);
    hipLaunchKernelGGL(k_merge, dim3((unsigned)(WM_ELEMS / 256)), dim3(256), 0, stream,
                       w, lA, lB, Wm);
    hipLaunchKernelGGL(k_gemm, dim3(MAX_DESC, OUT_D / M_TILE), dim3(256), 0, stream,
                       t, Wm, sorted, ctl, descs, out);
}